// VanillaLSTM_79937931313356
// MI455X (gfx1250) — compile-verified
//
#include <hip/hip_runtime.h>

// ---------------------------------------------------------------------------
// 2-layer LSTM (T=512, B=64, H=512) + Linear(512->128), fp32, CDNA5 WMMA.
// Persistent cooperative kernel: 32 WGs x 512 threads (16 wave32 each).
// Each wave computes one 16x16 gate tile per step with V_WMMA_F32_16X16X4_F32.
// Weights staged to LDS via GLOBAL_LOAD_ASYNC_TO_LDS_B128 when available.
// ---------------------------------------------------------------------------

typedef float v2f __attribute__((ext_vector_type(2)));
typedef float v8f __attribute__((ext_vector_type(8)));
typedef int   v4i __attribute__((ext_vector_type(4)));

#define T_STEPS   512
#define BATCH     64
#define HID       512
#define GATES     2048
#define OUTF      128
#define NWG       32
#define WG_THREADS 512
#define W_STRIDE  1028          // padded LDS row stride (floats) -> conflict-free b64 reads

// dynamic LDS partition (floats)
#define LDS_W_FLOATS (64 * W_STRIDE)            // 64 gate rows x 1024 K (padded)
#define LDS_G_OFF    LDS_W_FLOATS               // gbuf: [4 gate types][64 m][16 j]
#define LDS_C_OFF    (LDS_G_OFF + 4 * BATCH * 16)
#define LDS_TOTAL_FLOATS (LDS_C_OFF + BATCH * 16)   // + cbuf [64 m][16 j]

// gfx1250 async global->LDS copy path (ASYNCcnt), guarded per-toolchain
#if defined(__has_builtin)
#if __has_builtin(__builtin_amdgcn_global_load_async_to_lds_b128) && \
    __has_builtin(__builtin_amdgcn_s_wait_asynccnt)
#define USE_ASYNC_LDS 1
#endif
#endif
#ifndef USE_ASYNC_LDS
#define USE_ASYNC_LDS 0
#endif

__device__ __forceinline__ void stage_b128_to_lds(const float* src, float* dst_lds) {
#if USE_ASYNC_LDS
    // builtin expects pointers to 4x i32 vectors in AS(1)/AS(3)
    __attribute__((address_space(1))) void* gv =
        (__attribute__((address_space(1))) void*)src;
    __attribute__((address_space(3))) void* lv =
        (__attribute__((address_space(3))) void*)dst_lds;
    __builtin_amdgcn_global_load_async_to_lds_b128(
        (__attribute__((address_space(1))) v4i*)gv,
        (__attribute__((address_space(3))) v4i*)lv,
        0, 0);
#else
    float4 v = *(const float4*)src;
    dst_lds[0] = v.x; dst_lds[1] = v.y; dst_lds[2] = v.z; dst_lds[3] = v.w;
#endif
}

__device__ __forceinline__ float fast_sigmoid(float x) {
    // v_exp_f32 + v_rcp_f32 (avoid IEEE div expansion on the serial path)
    return __builtin_amdgcn_rcpf(1.0f + __expf(-x));
}
__device__ __forceinline__ float fast_tanh(float x) {
    return 2.0f * __builtin_amdgcn_rcpf(1.0f + __expf(-2.0f * x)) - 1.0f;
}

__device__ __forceinline__ void grid_barrier(int* cnt, int idx) {
    __syncthreads();
    if (threadIdx.x == 0) {
        __hip_atomic_fetch_add(&cnt[idx], 1, __ATOMIC_ACQ_REL, __HIP_MEMORY_SCOPE_AGENT);
        while (__hip_atomic_load(&cnt[idx], __ATOMIC_ACQUIRE, __HIP_MEMORY_SCOPE_AGENT) < NWG) {
            __builtin_amdgcn_s_sleep(1);
        }
    }
    __syncthreads();
    __threadfence();
}

__global__ void zero_counters(int* cnt, int n) {
    int i = blockIdx.x * blockDim.x + threadIdx.x;
    if (i < n) cnt[i] = 0;
}

__global__ __launch_bounds__(WG_THREADS)
void lstm2_persistent(const float* __restrict__ x,
                      const float* __restrict__ h0,
                      const float* __restrict__ c0,
                      const float* __restrict__ Wih1, const float* __restrict__ Whh1,
                      const float* __restrict__ bih1, const float* __restrict__ bhh1,
                      const float* __restrict__ Wih2, const float* __restrict__ Whh2,
                      const float* __restrict__ bih2, const float* __restrict__ bhh2,
                      const float* __restrict__ Wout, const float* __restrict__ bout,
                      float* __restrict__ hstate,   // ws: [64][512]
                      float* __restrict__ y1,       // ws: [T][64][512]
                      float* __restrict__ out,      // d_out: [64][128]
                      float* __restrict__ out_h2,   // d_out: [64][512]
                      float* __restrict__ out_c2,   // d_out: [64][512]
                      int* __restrict__ cnt)
{
    extern __shared__ float smem[];
    float* Wl   = smem;                 // [64][W_STRIDE]
    float* gbuf = smem + LDS_G_OFF;     // [4][64][16], post-activation gates
    float* cbuf = smem + LDS_C_OFF;     // [64][16], cell state for owned units

    const int wg   = blockIdx.x;        // [0,32): owns hidden cols [16*wg,16*wg+16)
    const int tid  = threadIdx.x;
    const int lane = tid & 31;
    const int wid  = tid >> 5;          // [0,16)
    const int gt   = wid >> 2;          // gate type: 0=i 1=f 2=g 3=o
    const int mt   = wid & 3;           // batch tile (16 rows)
    const int l15  = lane & 15;
    const int koff = (lane >> 4) * 2;   // K sub-offset for 16x16x4 f32 A/B operands

    // ---- init h-state / c-state for the columns this WG owns ----
    for (int e = tid; e < BATCH * 16; e += WG_THREADS) {
        int m = e >> 4, j = e & 15;
        int col = wg * 16 + j;
        hstate[m * HID + col] = h0[m * HID + col];
        cbuf[m * 16 + j]      = c0[m * HID + col];
    }
    int bar = 0;
    grid_barrier(cnt, bar++);

    for (int layer = 0; layer < 2; ++layer) {
        const float* Wih  = layer ? Wih2 : Wih1;
        const float* Whh  = layer ? Whh2 : Whh1;
        const float* bih  = layer ? bih2 : bih1;
        const float* bhh  = layer ? bhh2 : bhh1;
        const float* asrc = layer ? y1   : x;     // [T][64][512]

        // ---- stage 64x1024 weight strip into LDS: [W_ih | W_hh] ----
        for (int e = tid; e < 64 * 256; e += WG_THREADS) {
            int r  = e >> 8;                 // local gate row 0..63
            int cc = (e & 255) << 2;         // K col 0..1023 step 4
            int ng = (r >> 4) * 512 + wg * 16 + (r & 15);   // global gate row
            const float* src = (cc < HID) ? (Wih + (size_t)ng * HID + cc)
                                          : (Whh + (size_t)ng * HID + (cc - HID));
            stage_b128_to_lds(src, Wl + r * W_STRIDE + cc);
        }
#if USE_ASYNC_LDS
        __builtin_amdgcn_s_wait_asynccnt(0);
#endif
        const int ncol = gt * 512 + wg * 16 + l15;        // this lane's gate column
        const float bias = bih[ncol] + bhh[ncol];
        __syncthreads();

        for (int t = 0; t < T_STEPS; ++t) {
            // -------- gate GEMM: D = x_t*Wih^T + h*Whh^T (16x16 tile) --------
            const int m = mt * 16 + l15;                  // A row for this lane
            const float* arow_x = asrc + (size_t)t * BATCH * HID + (size_t)m * HID;
            const float* arow_h = hstate + (size_t)m * HID;
            const float* brow   = Wl + (gt * 16 + l15) * W_STRIDE;

            // speculative prefetch of next timestep's activations (global_prefetch_b8)
            __builtin_prefetch(arow_x + (size_t)BATCH * HID, 0, 0);

            v8f acc = {};
            #pragma unroll 8
            for (int kt = 0; kt < HID / 4; ++kt) {        // input-weight half (K 0..511)
                int k = kt * 4 + koff;
                v2f a = *(const v2f*)(arow_x + k);
                v2f b = *(const v2f*)(brow + k);
                acc = __builtin_amdgcn_wmma_f32_16x16x4_f32(
                          false, a, false, b, (short)0, acc, false, false);
            }
            #pragma unroll 8
            for (int kt = 0; kt < HID / 4; ++kt) {        // recurrent half (K 512..1023)
                int k = kt * 4 + koff;
                v2f a = *(const v2f*)(arow_h + k);
                v2f b = *(const v2f*)(brow + HID + k);
                acc = __builtin_amdgcn_wmma_f32_16x16x4_f32(
                          false, a, false, b, (short)0, acc, false, false);
            }

            // bias + activation, scatter to LDS gate buffer
            // D layout: lane -> n = lane&15, VGPR v -> m = v + 8*(lane>>4)
            float* gdst = gbuf + ((gt * BATCH) + mt * 16 + ((lane >> 4) << 3)) * 16 + l15;
            #pragma unroll
            for (int v = 0; v < 8; ++v) {
                float val = acc[v] + bias;
                val = (gt == 2) ? fast_tanh(val) : fast_sigmoid(val);
                gdst[v * 16] = val;
            }
            __syncthreads();

            // -------- elementwise cell update (WG-local hidden units) --------
            const bool last = (layer == 1) && (t == T_STEPS - 1);
            #pragma unroll
            for (int r = 0; r < 2; ++r) {
                int e  = tid * 2 + r;                     // [0,1024)
                int m2 = e >> 4, j = e & 15;
                float iv = gbuf[(0 * BATCH + m2) * 16 + j];
                float fv = gbuf[(1 * BATCH + m2) * 16 + j];
                float gv = gbuf[(2 * BATCH + m2) * 16 + j];
                float ov = gbuf[(3 * BATCH + m2) * 16 + j];
                float cnew = fv * cbuf[m2 * 16 + j] + iv * gv;
                float hnew = ov * fast_tanh(cnew);
                cbuf[m2 * 16 + j] = cnew;
                int col = wg * 16 + j;
                hstate[m2 * HID + col] = hnew;
                if (layer == 0)
                    y1[(size_t)t * BATCH * HID + m2 * HID + col] = hnew;
                if (last) {
                    out_h2[m2 * HID + col] = hnew;
                    out_c2[m2 * HID + col] = cnew;
                }
            }
            grid_barrier(cnt, bar++);     // publish h for next step's GEMM
        }
        // note: layer-2 (h,c) deliberately inherit layer-1 FINAL state (PyTorch quirk)
    }

    // -------- final linear: out = h2 @ Wout^T + bout  ([64x512]x[512x128]) --------
    // 32 tiles of 16x16 -> WGs 0 and 1, one tile per wave
    if (wg < 2) {
        int tile = wg * 16 + wid;         // [0,32)
        int tmt  = tile >> 3;             // m tile 0..3
        int tnt  = tile & 7;              // n tile 0..7
        int am   = tmt * 16 + l15;
        int bn   = tnt * 16 + l15;
        const float* arow = hstate + (size_t)am * HID;
        const float* brow = Wout + (size_t)bn * HID;
        v8f acc = {};
        #pragma unroll 8
        for (int kt = 0; kt < HID / 4; ++kt) {
            int k = kt * 4 + koff;
            v2f a = *(const v2f*)(arow + k);
            v2f b = *(const v2f*)(brow + k);
            acc = __builtin_amdgcn_wmma_f32_16x16x4_f32(
                      false, a, false, b, (short)0, acc, false, false);
        }
        float bo = bout[bn];
        #pragma unroll
        for (int v = 0; v < 8; ++v) {
            int mrow = tmt * 16 + v + ((lane >> 4) << 3);
            out[mrow * OUTF + bn] = acc[v] + bo;
        }
    }
}

extern "C" void kernel_launch(void* const* d_in, const int* in_sizes, int n_in,
                              void* d_out, int out_size, void* d_ws, size_t ws_size,
                              hipStream_t stream) {
    (void)in_sizes; (void)n_in; (void)out_size; (void)ws_size;

    const float* x    = (const float*)d_in[0];
    const float* h0   = (const float*)d_in[1];
    const float* c0   = (const float*)d_in[2];
    const float* Wih1 = (const float*)d_in[3];
    const float* Whh1 = (const float*)d_in[4];
    const float* bih1 = (const float*)d_in[5];
    const float* bhh1 = (const float*)d_in[6];
    const float* Wih2 = (const float*)d_in[7];
    const float* Whh2 = (const float*)d_in[8];
    const float* bih2 = (const float*)d_in[9];
    const float* bhh2 = (const float*)d_in[10];
    const float* Wout = (const float*)d_in[11];
    const float* bout = (const float*)d_in[12];

    float* out    = (float*)d_out;                 // [64][128]
    float* out_h2 = out + BATCH * OUTF;            // [1][64][512]
    float* out_c2 = out_h2 + BATCH * HID;          // [1][64][512]

    // workspace layout: counters (2048 ints) | hstate (64*512 f32) | y1 (T*64*512 f32)
    int*   cnt    = (int*)d_ws;
    float* hstate = (float*)((char*)d_ws + 2048 * sizeof(int));
    float* y1buf  = hstate + BATCH * HID;

    zero_counters<<<8, 256, 0, stream>>>(cnt, 2048);

    size_t lds_bytes = (size_t)LDS_TOTAL_FLOATS * sizeof(float);
    lstm2_persistent<<<NWG, WG_THREADS, lds_bytes, stream>>>(
        x, h0, c0, Wih1, Whh1, bih1, bhh1, Wih2, Whh2, bih2, bhh2,
        Wout, bout, hstate, y1buf, out, out_h2, out_c2, cnt);
}